// ProtoNetHead_79388175499456
// MI455X (gfx1250) — compile-verified
//
#include <hip/hip_runtime.h>
#include <hip/hip_bf16.h>

typedef __attribute__((ext_vector_type(2))) float v2f;
typedef __attribute__((ext_vector_type(8))) float v8f;

#define NS_   65536
#define NQ_   65536
#define DD_   512
#define NWAY_ 64

// ---------------- Kernel 1: zero accumulators ----------------
__global__ void k_zero(float* __restrict__ p, int n) {
  int i = blockIdx.x * blockDim.x + threadIdx.x;
  if (i < n) p[i] = 0.0f;
}

// ---------------- Kernel 2: segment-sum prototypes ----------------
// Each block handles 1024 support rows. D is chunked into 4 x 128 so the
// per-class LDS accumulator is 64*128*4B = 32KB. z_supp is read exactly once.
__global__ __launch_bounds__(256) void k_proto_accum(
    const float* __restrict__ zs, const int* __restrict__ ys,
    float* __restrict__ gcount, float* __restrict__ gsum) {
  __shared__ float sacc[NWAY_ * 128];
  __shared__ int   shist[NWAY_];
  const int tid = threadIdx.x;
  const int rowBase = blockIdx.x * 1024;

  // per-block class histogram -> global counts (as float)
  if (tid < NWAY_) shist[tid] = 0;
  __syncthreads();
  for (int r = tid; r < 1024; r += 256) atomicAdd(&shist[ys[rowBase + r]], 1);
  __syncthreads();
  if (tid < NWAY_) atomicAdd(&gcount[tid], (float)shist[tid]);

  const int col  = tid & 127;   // dim within chunk
  const int half = tid >> 7;    // two rows in flight per iteration
  for (int c = 0; c < 4; ++c) {
    for (int i = tid; i < NWAY_ * 128; i += 256) sacc[i] = 0.0f;
    __syncthreads();
    for (int r0 = 0; r0 < 1024; r0 += 2) {
      const int row = rowBase + r0 + half;
      const int y = ys[row];
      const float v = zs[(size_t)row * DD_ + c * 128 + col];
      atomicAdd(&sacc[y * 128 + col], v);   // ds_add_f32, conflict-free across dims
    }
    __syncthreads();
    for (int i = tid; i < NWAY_ * 128; i += 256) {
      const int y = i >> 7, d = i & 127;
      atomicAdd(&gsum[(size_t)y * DD_ + c * 128 + d], sacc[i]);
    }
    __syncthreads();
  }
}

// ---------------- Kernel 3: finalize prototypes ----------------
// protos[p][d] = sum/count; store in WMMA-B paired layout:
//   PB float2 element [(d/2)*64 + p] = (proto[p][d_even], proto[p][d_even+1])
// Also compute b2[p] = ||proto_p||^2 and emit the classes output.
__global__ __launch_bounds__(512) void k_finalize(
    const float* __restrict__ gsum, const float* __restrict__ gcount,
    float* __restrict__ PB, float* __restrict__ b2, int* __restrict__ classes_out) {
  __shared__ float red[512];
  const int p = blockIdx.x;
  const int d = threadIdx.x;
  const float cnt = fmaxf(gcount[p], 1.0f);
  const float v = gsum[(size_t)p * DD_ + d] / cnt;
  PB[(size_t)(d >> 1) * (2 * NWAY_) + p * 2 + (d & 1)] = v;
  red[d] = v * v;
  __syncthreads();
  for (int s = 256; s > 0; s >>= 1) {
    if (d < s) red[d] += red[d + s];
    __syncthreads();
  }
  if (d == 0) b2[p] = red[0];
  if (p == 0 && d < NWAY_) classes_out[d] = d;
}

// ---------------- Kernel 4: WMMA GEMM + distance epilogue ----------------
// 256 threads = 8 wave32; each wave owns a 16-row x 64-col logits tile
// (4x v8f accumulators). K loop: 128 steps of v_wmma_f32_16x16x4_f32 per N-tile.
__global__ __launch_bounds__(256) void k_gemm(
    const float* __restrict__ zq, const float* __restrict__ PB,
    const float* __restrict__ b2, const float* __restrict__ log_temp,
    float* __restrict__ out) {
  const int tid  = threadIdx.x;
  const int lane = tid & 31;
  const int wave = tid >> 5;
  const int l15  = lane & 15;   // A: row M in tile / B: col N in n-tile
  const int h    = lane >> 4;   // half-wave selects K pair (0 -> k0..1, 1 -> k2..3)

  const long long qBase = (long long)blockIdx.x * 128 + wave * 16;
  const float* aRow = zq + (qBase + l15) * DD_;

  v8f acc0 = {}, acc1 = {}, acc2 = {}, acc3 = {};
  float a2p = 0.0f;

#pragma unroll 4
  for (int k = 0; k < DD_; k += 4) {
    const int kp = (k >> 1) + h;  // float2 pair index into PB
    const v2f a = *(const v2f*)(aRow + k + 2 * h);
    a2p += a.x * a.x + a.y * a.y;
    const v2f b0 = *(const v2f*)(PB + 2 * ((size_t)kp * NWAY_ +  0 + l15));
    const v2f b1 = *(const v2f*)(PB + 2 * ((size_t)kp * NWAY_ + 16 + l15));
    const v2f b2f = *(const v2f*)(PB + 2 * ((size_t)kp * NWAY_ + 32 + l15));
    const v2f b3 = *(const v2f*)(PB + 2 * ((size_t)kp * NWAY_ + 48 + l15));
    acc0 = __builtin_amdgcn_wmma_f32_16x16x4_f32(false, a, false, b0, (short)0, acc0, false, false);
    acc1 = __builtin_amdgcn_wmma_f32_16x16x4_f32(false, a, false, b1, (short)0, acc1, false, false);
    acc2 = __builtin_amdgcn_wmma_f32_16x16x4_f32(false, a, false, b2f, (short)0, acc2, false, false);
    acc3 = __builtin_amdgcn_wmma_f32_16x16x4_f32(false, a, false, b3, (short)0, acc3, false, false);
  }

  // ||q||^2: lanes L and L^16 jointly cover all K for row L%16 -> combine,
  // then broadcast row m from lane m with ds_bpermute (no LDS memory).
  const float a2full = a2p + __int_as_float(
      __builtin_amdgcn_ds_bpermute((lane ^ 16) << 2, __float_as_int(a2p)));

  const float invT = 1.0f / fmaxf(expf(log_temp[0]), 1e-6f);
  const float p0 = b2[ 0 + l15];
  const float p1 = b2[16 + l15];
  const float p2 = b2[32 + l15];
  const float p3 = b2[48 + l15];

#pragma unroll
  for (int r = 0; r < 8; ++r) {
    const int rowInTile = r + 8 * h;  // C/D layout: VGPR r, lane half selects +8
    const float a2row = __int_as_float(
        __builtin_amdgcn_ds_bpermute(rowInTile << 2, __float_as_int(a2full)));
    float* orow = out + (qBase + rowInTile) * NWAY_;
    orow[ 0 + l15] = (2.0f * acc0[r] - a2row - p0) * invT;
    orow[16 + l15] = (2.0f * acc1[r] - a2row - p1) * invT;
    orow[32 + l15] = (2.0f * acc2[r] - a2row - p2) * invT;
    orow[48 + l15] = (2.0f * acc3[r] - a2row - p3) * invT;
  }
}

extern "C" void kernel_launch(void* const* d_in, const int* in_sizes, int n_in,
                              void* d_out, int out_size, void* d_ws, size_t ws_size,
                              hipStream_t stream) {
  const float* z_supp   = (const float*)d_in[0];
  const float* z_query  = (const float*)d_in[1];
  const float* log_temp = (const float*)d_in[2];
  const int*   y_supp   = (const int*)d_in[3];
  (void)in_sizes; (void)n_in; (void)out_size; (void)ws_size;

  // Workspace layout (floats): [counts 64][sums 64*512][PB 512*64][b2 64]
  float* ws     = (float*)d_ws;
  float* gcount = ws;
  float* gsum   = ws + 64;
  float* PB     = gsum + NWAY_ * DD_;
  float* b2v    = PB + (size_t)DD_ * NWAY_;

  float* out = (float*)d_out;
  int* classes_out = (int*)d_out + (long long)NQ_ * NWAY_;

  const int nz = 64 + NWAY_ * DD_;
  k_zero<<<(nz + 255) / 256, 256, 0, stream>>>(ws, nz);
  k_proto_accum<<<NS_ / 1024, 256, 0, stream>>>(z_supp, y_supp, gcount, gsum);
  k_finalize<<<NWAY_, 512, 0, stream>>>(gsum, gcount, PB, b2v, classes_out);
  k_gemm<<<NQ_ / 128, 256, 0, stream>>>(z_query, PB, b2v, log_temp, out);
}